// EDLayer_47450798686392
// MI455X (gfx1250) — compile-verified
//
#include <hip/hip_runtime.h>
#include <hip/hip_bf16.h>

typedef float v2f __attribute__((ext_vector_type(2)));
typedef float v8f __attribute__((ext_vector_type(8)));

#define MASK_VALUE (-100.0f)
#define TOPK   5
#define C_CAND 30
#define L_TOK  32
#define NCLS   64
#define D_IN   768
#define D_OUT  300

// ---------------------------------------------------------------------------
// Kernel 1: m = A[ N x 768 ] @ W[ 768 x 300 ] + b  via V_WMMA_F32_16X16X4_F32
// One wave per 16x32 output tile (2 adjacent N-tiles share each A fragment).
// grid = (10 n-tile-pairs, N/16). Column indices are clamped for loads so the
// inner loop has NO divergence (no exec-mask save/restore); the store guard
// handles the partial last tile (300 = 9*32 + 12).
//
// A-frag layout (32-bit A 16x4): lanes 0-15 -> M=lane,   K={k,k+1};
//                                lanes 16-31 -> M=lane-16, K={k+2,k+3}.
// B-frag layout (32-bit B 4x16): VGPR0: K={k(+2)} rows, VGPR1: K={k+1(+3)}.
// C/D layout: VGPR r, lane l -> M = r + (l>>4)*8, N = l&15.
// ---------------------------------------------------------------------------
__global__ __launch_bounds__(32)
void edlayer_proj_wmma(const float* __restrict__ A,
                       const float* __restrict__ W,
                       const float* __restrict__ bias,
                       float* __restrict__ Mout) {
    const int ntile = blockIdx.x;            // pair of 16-wide N tiles
    const int mtile = blockIdx.y;
    const int lane  = threadIdx.x;

    const int mrow  = mtile * 16 + (lane & 15);
    const int ncol0 = ntile * 32 + (lane & 15);
    const int ncol1 = ncol0 + 16;
    const int kb    = (lane >> 4) * 2;       // K sub-offset for this half-wave

    // clamped (always-valid) load columns; store guard does real bounds check
    const int nc0 = (ncol0 < D_OUT) ? ncol0 : (D_OUT - 1);
    const int nc1 = (ncol1 < D_OUT) ? ncol1 : (D_OUT - 1);

    const float bv0 = bias[nc0];
    const float bv1 = bias[nc1];
    v8f acc0, acc1;
#pragma unroll
    for (int r = 0; r < 8; ++r) { acc0[r] = bv0; acc1[r] = bv1; }

    const float* __restrict__ arow = A + (size_t)mrow * D_IN + kb;
    const float* __restrict__ w0   = W + (size_t)kb * D_OUT + nc0;
    const float* __restrict__ w1   = W + (size_t)kb * D_OUT + nc1;

#pragma unroll 4
    for (int k = 0; k < D_IN; k += 4) {
        v2f a, b0, b1;
        a.x  = arow[k];
        a.y  = arow[k + 1];
        b0.x = w0[(size_t)k * D_OUT];
        b0.y = w0[(size_t)(k + 1) * D_OUT];
        b1.x = w1[(size_t)k * D_OUT];
        b1.y = w1[(size_t)(k + 1) * D_OUT];
        acc0 = __builtin_amdgcn_wmma_f32_16x16x4_f32(
            false, a, false, b0, (short)0, acc0, false, false);
        acc1 = __builtin_amdgcn_wmma_f32_16x16x4_f32(
            false, a, false, b1, (short)0, acc1, false, false);
    }

    const int mbase = mtile * 16 + (lane >> 4) * 8;
    if (ncol0 < D_OUT) {
#pragma unroll
        for (int r = 0; r < 8; ++r)
            Mout[(size_t)(mbase + r) * D_OUT + ncol0] = acc0[r];
    }
    if (ncol1 < D_OUT) {
#pragma unroll
        for (int r = 0; r < 8; ++r)
            Mout[(size_t)(mbase + r) * D_OUT + ncol1] = acc1[r];
    }
}

// ---------------------------------------------------------------------------
// Kernel 2: scores, mask, gold redirect, top-5, gathers. One block per row.
// ---------------------------------------------------------------------------
__global__ __launch_bounds__(256)
void edlayer_score_topk(const float* __restrict__ m_ws,
                        const int*   __restrict__ desc,       // [N,C,L]
                        const float* __restrict__ pem,        // [N,C]
                        const float* __restrict__ classes,    // [N,C,NCLS]
                        const float* __restrict__ ent_tgt,    // [N,C]
                        const float* __restrict__ desc_emb,   // [N,C,D_OUT]
                        float* __restrict__ out,
                        int N) {
    __shared__ float m_s[D_OUT];
    __shared__ float sc_s[C_CAND];
    __shared__ int   sel_s[TOPK];
    __shared__ float rt_s;

    const int n   = blockIdx.x;
    const int tid = threadIdx.x;

    for (int d = tid; d < D_OUT; d += 256)
        m_s[d] = m_ws[(size_t)n * D_OUT + d];
    __syncthreads();

    const int wave = tid >> 5;
    const int lane = tid & 31;
    const float* __restrict__ de_row = desc_emb + (size_t)n * C_CAND * D_OUT;

    for (int c = wave; c < C_CAND; c += 8) {
        const float* __restrict__ v = de_row + (size_t)c * D_OUT;
        float p = 0.0f;
        for (int d = lane; d < D_OUT; d += 32)
            p += v[d] * m_s[d];
#pragma unroll
        for (int off = 16; off >= 1; off >>= 1)
            p += __shfl_xor(p, off, 32);
        if (lane == 0) {
            const bool mult = (v[0] != 0.0f);
            sc_s[c] = mult ? p : MASK_VALUE;
        }
    }
    __syncthreads();

    if (tid == 0) {
        // argmax of candidate_entity_targets (first max wins, as jnp.argmax)
        const float* t = ent_tgt + (size_t)n * C_CAND;
        int tgt = 0; float tv = t[0];
        for (int c = 1; c < C_CAND; ++c)
            if (t[c] > tv) { tv = t[c]; tgt = c; }

        const float gold_score = sc_s[tgt];
        const bool  has_gold   = (gold_score != MASK_VALUE);
        const int   safe_gold  = has_gold ? tgt : (C_CAND - 1);

        // top-5 by repeated max (strict > => lower index wins ties, like lax.top_k)
        int  top_idx[TOPK];
        bool used[C_CAND];
        for (int c = 0; c < C_CAND; ++c) used[c] = false;
        for (int j = 0; j < TOPK; ++j) {
            int bi = 0; float bvv = -1.0e30f;
            for (int c = 0; c < C_CAND; ++c)
                if (!used[c] && sc_s[c] > bvv) { bvv = sc_s[c]; bi = c; }
            used[bi]   = true;
            top_idx[j] = bi;
        }

        int  pos = 0; bool found = false;
        for (int j = 0; j < TOPK; ++j)
            if (!found && top_idx[j] == safe_gold) { pos = j; found = true; }
        const bool in_topk     = has_gold && found;
        const bool use_replace = has_gold && !in_topk;

        for (int j = 0; j < TOPK; ++j) sel_s[j] = top_idx[j];
        if (use_replace) sel_s[TOPK - 1] = safe_gold;

        const int rt = (!has_gold) ? TOPK : (in_topk ? pos : (TOPK - 1));
        rt_s = (float)rt;
    }
    __syncthreads();

    // flat-concat output offsets (all as float)
    const size_t off_scores  = 0;
    const size_t off_targets = (size_t)N * TOPK;
    const size_t off_desc    = off_targets + (size_t)N;
    const size_t off_demb    = off_desc + (size_t)N * TOPK * L_TOK;
    const size_t off_pem     = off_demb + (size_t)N * TOPK * D_OUT;
    const size_t off_cls     = off_pem  + (size_t)N * TOPK;

    if (tid < TOPK) {
        const int s = sel_s[tid];
        out[off_scores + (size_t)n * TOPK + tid] = sc_s[s];
        out[off_pem    + (size_t)n * TOPK + tid] = pem[(size_t)n * C_CAND + s];
    }
    if (tid == 0)
        out[off_targets + (size_t)n] = rt_s;

    for (int i = tid; i < TOPK * L_TOK; i += 256) {
        const int j = i >> 5, l = i & (L_TOK - 1);
        out[off_desc + (size_t)n * TOPK * L_TOK + i] =
            (float)desc[((size_t)n * C_CAND + sel_s[j]) * L_TOK + l];
    }
    for (int i = tid; i < TOPK * D_OUT; i += 256) {
        const int j = i / D_OUT, d2 = i - j * D_OUT;
        out[off_demb + (size_t)n * TOPK * D_OUT + i] =
            de_row[(size_t)sel_s[j] * D_OUT + d2];
    }
    for (int i = tid; i < TOPK * NCLS; i += 256) {
        const int j = i >> 6, kk = i & (NCLS - 1);
        out[off_cls + (size_t)n * TOPK * NCLS + i] =
            classes[((size_t)n * C_CAND + sel_s[j]) * NCLS + kk];
    }
}

extern "C" void kernel_launch(void* const* d_in, const int* in_sizes, int n_in,
                              void* d_out, int out_size, void* d_ws, size_t ws_size,
                              hipStream_t stream) {
    const float* mention  = (const float*)d_in[0];   // [N, 768]
    const int*   desc     = (const int*)  d_in[1];   // [N, 30, 32]
    const float* pem      = (const float*)d_in[2];   // [N, 30]
    const float* classes  = (const float*)d_in[3];   // [N, 30, 64]
    const float* ent_tgt  = (const float*)d_in[4];   // [N, 30]
    const float* desc_emb = (const float*)d_in[5];   // [N, 30, 300]
    const float* W_proj   = (const float*)d_in[6];   // [768, 300]
    const float* b_proj   = (const float*)d_in[7];   // [300]

    const int N = in_sizes[0] / D_IN;                // 8192
    float* m_ws = (float*)d_ws;                      // [N, 300] scratch

    // Projection GEMM via f32 WMMA: 10 n-tile-pairs x N/16 m-tiles, 1 wave each.
    dim3 grid1((D_OUT + 31) / 32, (N + 15) / 16, 1);
    edlayer_proj_wmma<<<grid1, 32, 0, stream>>>(mention, W_proj, b_proj, m_ws);

    // Scores + top-k + gathers: one 256-thread block per row.
    edlayer_score_topk<<<N, 256, 0, stream>>>(m_ws, desc, pem, classes,
                                              ent_tgt, desc_emb,
                                              (float*)d_out, N);
    (void)n_in; (void)out_size; (void)ws_size;
}